// STEM_v2_origin_ps_50165218018142
// MI455X (gfx1250) — compile-verified
//
#include <hip/hip_runtime.h>
#include <hip/hip_bf16.h>

// ---------------------------------------------------------------------------
// Problem constants (from reference)
// ---------------------------------------------------------------------------
#define BB    64
#define CIN   3
#define JJ    11
#define TT    512
#define COUT  128
#define NTOT  (BB * JJ * TT)        // 360448 = 1408*256 = 88*4096 = 5632*64
#define JT    (JJ * TT)             // 5632
#define SIGC  84
#define K1    (2 * SIGC)            // 168
#define K2    (2 * COUT)            // 256
#define BN_EPS 1e-5f

typedef __attribute__((ext_vector_type(2))) float v2f;
typedef __attribute__((ext_vector_type(8))) float v8f;

// ---------------------------------------------------------------------------
// Depth-3 signature step (c=4), in-place (S3 first, then S2, then S1)
//   S3 += S2 (x) d + S1 (x) e2 + d(x)d(x)d/6,  e2 = 0.5 d(x)d
// ---------------------------------------------------------------------------
__device__ __forceinline__ void sig_step(float* S1, float* S2, float* S3,
                                         const float d[4]) {
    float e2[16];
#pragma unroll
    for (int i = 0; i < 4; ++i)
#pragma unroll
        for (int j = 0; j < 4; ++j) e2[i * 4 + j] = 0.5f * d[i] * d[j];
#pragma unroll
    for (int i = 0; i < 4; ++i) {
        const float s1e = S1[i] + d[i] * (1.0f / 3.0f);  // (S1_i + d_i/3)*e2 == S1(x)e2 + e3
#pragma unroll
        for (int j = 0; j < 4; ++j) {
            const float s2ij = S2[i * 4 + j];
#pragma unroll
            for (int k = 0; k < 4; ++k)
                S3[(i * 4 + j) * 4 + k] += s2ij * d[k] + s1e * e2[j * 4 + k];
        }
    }
#pragma unroll
    for (int i = 0; i < 4; ++i)
#pragma unroll
        for (int j = 0; j < 4; ++j) S2[i * 4 + j] += S1[i] * d[j] + e2[i * 4 + j];
#pragma unroll
    for (int i = 0; i < 4; ++i) S1[i] += d[i];
}

__device__ __forceinline__ void sig_clear(float* S1, float* S2, float* S3) {
#pragma unroll
    for (int i = 0; i < 4; ++i) S1[i] = 0.f;
#pragma unroll
    for (int i = 0; i < 16; ++i) S2[i] = 0.f;
#pragma unroll
    for (int i = 0; i < 64; ++i) S3[i] = 0.f;
}

// Write 84 signature channels to channel-major SIG[(base+c)*N + n]
__device__ __forceinline__ void sig_write(float* sig, int base, int n,
                                          const float* S1, const float* S2,
                                          const float* S3) {
#pragma unroll
    for (int i = 0; i < 4; ++i)  sig[(size_t)(base + i) * NTOT + n] = S1[i];
#pragma unroll
    for (int i = 0; i < 16; ++i) sig[(size_t)(base + 4 + i) * NTOT + n] = S2[i];
#pragma unroll
    for (int i = 0; i < 64; ++i) sig[(size_t)(base + 20 + i) * NTOT + n] = S3[i];
}

// ---------------------------------------------------------------------------
// Kernel: zero the BN-stat accumulators (768 floats)
// ---------------------------------------------------------------------------
__global__ void zero_stats_k(float* stats) {
    for (int i = threadIdx.x; i < 6 * COUT; i += blockDim.x) stats[i] = 0.f;
}

// ---------------------------------------------------------------------------
// Kernel: spatial + temporal depth-3 signatures -> SIG [168][N] channel-major
// ---------------------------------------------------------------------------
__global__ void signature_k(const float* __restrict__ x,
                            const int* __restrict__ path,
                            float* __restrict__ sig) {
    const int n = blockIdx.x * blockDim.x + threadIdx.x;  // exact: 1408*256
    const int b = n / JT;
    const int jt = n - b * JT;
    const int j = jt / TT;
    const int t = jt - j * TT;

    float S1[4], S2[16], S3[64], d[4];

    // ---- spatial path: 3 joints, time increments {0, 0.5, 0.5} ----
    sig_clear(S1, S2, S3);
    const int ja = path[j * 3 + 0], jb = path[j * 3 + 1], jc = path[j * 3 + 2];
    float pa[3], pb[3], pc[3];
#pragma unroll
    for (int c = 0; c < CIN; ++c) {
        const size_t base = ((size_t)(b * CIN + c)) * JJ;
        pa[c] = x[(base + ja) * TT + t];
        pb[c] = x[(base + jb) * TT + t];
        pc[c] = x[(base + jc) * TT + t];
    }
    d[0] = 0.f;   d[1] = pa[0];          d[2] = pa[1];          d[3] = pa[2];
    sig_step(S1, S2, S3, d);
    d[0] = 0.5f;  d[1] = pb[0] - pa[0];  d[2] = pb[1] - pa[1];  d[3] = pb[2] - pa[2];
    sig_step(S1, S2, S3, d);
    d[0] = 0.5f;  d[1] = pc[0] - pb[0];  d[2] = pc[1] - pb[1];  d[3] = pc[2] - pb[2];
    sig_step(S1, S2, S3, d);
    sig_write(sig, 0, n, S1, S2, S3);

    // ---- temporal path: 7 edge-clamped steps, time increments {0, 1/6 x6} ----
    sig_clear(S1, S2, S3);
    float prev[3], cur[3];
    {
        const int tc = (t - 3 < 0) ? 0 : (t - 3);
#pragma unroll
        for (int c = 0; c < CIN; ++c)
            prev[c] = x[(((size_t)(b * CIN + c)) * JJ + j) * TT + tc];
    }
    d[0] = 0.f; d[1] = prev[0]; d[2] = prev[1]; d[3] = prev[2];
    sig_step(S1, S2, S3, d);
#pragma unroll
    for (int l = 1; l < 7; ++l) {
        int tc = t - 3 + l;
        tc = tc < 0 ? 0 : (tc > TT - 1 ? TT - 1 : tc);
#pragma unroll
        for (int c = 0; c < CIN; ++c)
            cur[c] = x[(((size_t)(b * CIN + c)) * JJ + j) * TT + tc];
        d[0] = 1.0f / 6.0f;
        d[1] = cur[0] - prev[0]; d[2] = cur[1] - prev[1]; d[3] = cur[2] - prev[2];
        sig_step(S1, S2, S3, d);
        prev[0] = cur[0]; prev[1] = cur[1]; prev[2] = cur[2];
    }
    sig_write(sig, SIGC, n, S1, S2, S3);
}

// ---------------------------------------------------------------------------
// Kernel: raw 1x3 temporal conv (zero-padded) -> PRE1 [128][N] channel-major
// (bias omitted: it cancels exactly inside BatchNorm)
// ---------------------------------------------------------------------------
__global__ void raw_conv_k(const float* __restrict__ x,
                           const float* __restrict__ w,  // [128][3][1][3]
                           float* __restrict__ pre1) {
    __shared__ float wl[COUT * 9];
    for (int i = threadIdx.x; i < COUT * 9; i += blockDim.x) wl[i] = w[i];
    __syncthreads();

    const int n = blockIdx.x * blockDim.x + threadIdx.x;  // exact grid
    const int b = n / JT;
    const int jt = n - b * JT;
    const int j = jt / TT;
    const int t = jt - j * TT;

    float xv[9];
#pragma unroll
    for (int ci = 0; ci < CIN; ++ci)
#pragma unroll
        for (int kk = 0; kk < 3; ++kk) {
            const int tt = t + kk - 1;
            xv[ci * 3 + kk] = (tt >= 0 && tt < TT)
                ? x[(((size_t)(b * CIN + ci)) * JJ + j) * TT + tt] : 0.f;
        }
    for (int o = 0; o < COUT; ++o) {
        float acc = 0.f;
#pragma unroll
        for (int q = 0; q < 9; ++q) acc += wl[o * 9 + q] * xv[q];
        pre1[(size_t)o * NTOT + n] = acc;
    }
}

// ---------------------------------------------------------------------------
// Kernel: fp32 WMMA GEMM   D[128][N] = A[128][K] * B[K][N]
// 256 threads = 8 waves; wave w owns a 16-channel x 64-column tile
// (4 accumulators). One A fragment is reused by 4 independent
// V_WMMA_F32_16X16X4_F32 ops per k-step. No divergence (EXEC all ones).
// ---------------------------------------------------------------------------
__global__ void gemm_wmma_k(const float* __restrict__ A,
                            const float* __restrict__ Bm,
                            float* __restrict__ D, int K) {
    const int lane = threadIdx.x & 31;
    const int wave = threadIdx.x >> 5;
    const int half = lane >> 4;      // 0: K rows 0..1 | 1: K rows 2..3
    const int ln   = lane & 15;
    const int o_base = wave * 16;
    const size_t n_base = (size_t)blockIdx.x * 64;

    v8f acc0 = {}, acc1 = {}, acc2 = {}, acc3 = {};
    const float* Arow = A + (size_t)(o_base + ln) * K;
    for (int k0 = 0; k0 < K; k0 += 4) {
        const int ka = k0 + 2 * half;
        v2f a;
        a.x = Arow[ka];
        a.y = Arow[ka + 1];
        const float* Bp = Bm + (size_t)ka * NTOT + n_base + ln;
        v2f b0, b1, b2, b3;
        b0.x = Bp[0];   b0.y = Bp[NTOT + 0];
        b1.x = Bp[16];  b1.y = Bp[NTOT + 16];
        b2.x = Bp[32];  b2.y = Bp[NTOT + 32];
        b3.x = Bp[48];  b3.y = Bp[NTOT + 48];
        acc0 = __builtin_amdgcn_wmma_f32_16x16x4_f32(false, a, false, b0,
                                                     (short)0, acc0, false, false);
        acc1 = __builtin_amdgcn_wmma_f32_16x16x4_f32(false, a, false, b1,
                                                     (short)0, acc1, false, false);
        acc2 = __builtin_amdgcn_wmma_f32_16x16x4_f32(false, a, false, b2,
                                                     (short)0, acc2, false, false);
        acc3 = __builtin_amdgcn_wmma_f32_16x16x4_f32(false, a, false, b3,
                                                     (short)0, acc3, false, false);
    }
    // D layout: VGPR r holds row (r + 8*half), col = ln (+16 per n-subtile)
    float* Dp = D + (size_t)(o_base + 8 * half) * NTOT + n_base + ln;
#pragma unroll
    for (int r = 0; r < 8; ++r) {
        float* q = Dp + (size_t)r * NTOT;
        q[0]  = acc0[r];
        q[16] = acc1[r];
        q[32] = acc2[r];
        q[48] = acc3[r];
    }
}

// ---------------------------------------------------------------------------
// Kernel: per-channel sum / sum-of-squares over N (for BN mean/var)
// grid (128, 88), 4096 elements per block, LDS reduce + global float atomics.
// ---------------------------------------------------------------------------
__global__ void stats_k(const float* __restrict__ data,
                        float* __restrict__ sum, float* __restrict__ sq) {
    __shared__ float ssum[256], ssq[256];
    const int c = blockIdx.x;
    const float* p = data + (size_t)c * NTOT + (size_t)blockIdx.y * 4096;
    float s = 0.f, q = 0.f;
    for (int i = threadIdx.x; i < 4096; i += 256) {
        const float v = p[i];
        s += v; q += v * v;
    }
    ssum[threadIdx.x] = s; ssq[threadIdx.x] = q;
    __syncthreads();
    for (int off = 128; off > 0; off >>= 1) {
        if (threadIdx.x < off) {
            ssum[threadIdx.x] += ssum[threadIdx.x + off];
            ssq[threadIdx.x]  += ssq[threadIdx.x + off];
        }
        __syncthreads();
    }
    if (threadIdx.x == 0) {
        atomicAdd(&sum[c], ssum[0]);
        atomicAdd(&sq[c],  ssq[0]);
    }
}

// ---------------------------------------------------------------------------
// Kernel: in-place BN (biased var) + ReLU on a [128][N] tensor
// ---------------------------------------------------------------------------
__global__ void bn_relu_k(float* __restrict__ data,
                          const float* __restrict__ sum,
                          const float* __restrict__ sq,
                          const float* __restrict__ gamma,
                          const float* __restrict__ beta) {
    const int idx = blockIdx.x * blockDim.x + threadIdx.x;  // exact grid
    const int c = idx / NTOT;
    const float invN = 1.0f / (float)NTOT;
    const float mean = sum[c] * invN;
    const float var  = sq[c] * invN - mean * mean;
    const float scale = gamma[c] * rsqrtf(var + BN_EPS);
    const float v = (data[idx] - mean) * scale + beta[c];
    data[idx] = v > 0.f ? v : 0.f;
}

// ---------------------------------------------------------------------------
// Kernel: final BN + ReLU, permute [c][n=(b,j,t)] -> out[b][c][j][t]
// ---------------------------------------------------------------------------
__global__ void bn_relu_out_k(const float* __restrict__ data,
                              const float* __restrict__ sum,
                              const float* __restrict__ sq,
                              const float* __restrict__ gamma,
                              const float* __restrict__ beta,
                              float* __restrict__ out) {
    const int idx = blockIdx.x * blockDim.x + threadIdx.x;  // exact grid
    const int c = idx / NTOT;
    const int n = idx - c * NTOT;
    const int b = n / JT;
    const int r = n - b * JT;  // j*T + t
    const float invN = 1.0f / (float)NTOT;
    const float mean = sum[c] * invN;
    const float var  = sq[c] * invN - mean * mean;
    const float scale = gamma[c] * rsqrtf(var + BN_EPS);
    const float v = (data[idx] - mean) * scale + beta[c];
    out[((size_t)(b * COUT + c)) * JT + r] = v > 0.f ? v : 0.f;
}

// ---------------------------------------------------------------------------
// Host orchestration
// ---------------------------------------------------------------------------
extern "C" void kernel_launch(void* const* d_in, const int* in_sizes, int n_in,
                              void* d_out, int out_size, void* d_ws, size_t ws_size,
                              hipStream_t stream) {
    const float* x        = (const float*)d_in[0];
    const int*   path     = (const int*)  d_in[1];
    const float* raw_w    = (const float*)d_in[2];
    const float* raw_g    = (const float*)d_in[4];
    const float* raw_beta = (const float*)d_in[5];
    const float* ps_w     = (const float*)d_in[6];
    const float* ps_g     = (const float*)d_in[8];
    const float* ps_beta  = (const float*)d_in[9];
    const float* fus_w    = (const float*)d_in[10];
    const float* fus_g    = (const float*)d_in[12];
    const float* fus_beta = (const float*)d_in[13];
    float* out = (float*)d_out;

    // Workspace layout (fp32):
    //   SIG  [168][N]  (later reused for PRE3 [128][N])
    //   PRE1 [128][N]  -+ contiguous: forms the [256][N] input of the fused GEMM
    //   PRE2 [128][N]  -+
    //   STATS: 6*128 floats  (sum1,sq1,sum2,sq2,sum3,sq3)
    float* SIG   = (float*)d_ws;
    float* PRE1  = SIG  + (size_t)K1   * NTOT;
    float* PRE2  = PRE1 + (size_t)COUT * NTOT;
    float* STATS = PRE2 + (size_t)COUT * NTOT;
    float* sum1 = STATS + 0 * COUT, * sq1 = STATS + 1 * COUT;
    float* sum2 = STATS + 2 * COUT, * sq2 = STATS + 3 * COUT;
    float* sum3 = STATS + 4 * COUT, * sq3 = STATS + 5 * COUT;
    float* PRE3 = SIG;  // reuse (128N <= 168N)

    const dim3 blk(256);
    const int  nThreadBlks = NTOT / 256;            // 1408 (exact)
    const dim3 statsGrid(COUT, NTOT / 4096);        // (128, 88) exact
    const int  elemBlks = (COUT * NTOT) / 256;      // 180224 (exact)
    const int  gemmBlks = NTOT / 64;                // 5632 (exact)

    zero_stats_k<<<1, blk, 0, stream>>>(STATS);

    signature_k<<<nThreadBlks, blk, 0, stream>>>(x, path, SIG);
    raw_conv_k <<<nThreadBlks, blk, 0, stream>>>(x, raw_w, PRE1);

    // GEMM1: PRE2 = ps_w[128][168] * SIG[168][N]
    gemm_wmma_k<<<gemmBlks, blk, 0, stream>>>(ps_w, SIG, PRE2, K1);

    stats_k<<<statsGrid, blk, 0, stream>>>(PRE1, sum1, sq1);
    stats_k<<<statsGrid, blk, 0, stream>>>(PRE2, sum2, sq2);

    bn_relu_k<<<elemBlks, blk, 0, stream>>>(PRE1, sum1, sq1, raw_g, raw_beta);
    bn_relu_k<<<elemBlks, blk, 0, stream>>>(PRE2, sum2, sq2, ps_g, ps_beta);

    // GEMM2: PRE3 = fus_w[128][256] * concat(PRE1,PRE2)[256][N]
    gemm_wmma_k<<<gemmBlks, blk, 0, stream>>>(fus_w, PRE1, PRE3, K2);

    stats_k<<<statsGrid, blk, 0, stream>>>(PRE3, sum3, sq3);

    bn_relu_out_k<<<elemBlks, blk, 0, stream>>>(PRE3, sum3, sq3, fus_g, fus_beta, out);
}